// SCHBR_16750372454912
// MI455X (gfx1250) — compile-verified
//
#include <hip/hip_runtime.h>

#define NUM_USERS   100000
#define NUM_BUNDLES 50000
#define N_NODES     150000   // NUM_USERS + NUM_BUNDLES
#define EMB         64
#define NNZ         3000000
#define BATCH       2048
#define NUM_CAND    100
#define L2NORM      0.0001f

typedef __attribute__((ext_vector_type(2))) float v2f;
typedef __attribute__((ext_vector_type(8))) float v8f;

// ---------------------------------------------------------------------------
// Kernel 1: acc = 0.5 * embed_0  (virtual concat of users_feature|bundles_feature)
// also zero the loss scalar in d_out.
// ---------------------------------------------------------------------------
__global__ void k_init(const float* __restrict__ uf, const float* __restrict__ bf,
                       float* __restrict__ acc, float* __restrict__ loss) {
    const int n4 = N_NODES * EMB / 4;          // 2,400,000 float4s
    const int u4 = NUM_USERS * EMB / 4;        // 1,600,000 float4s
    int gid = blockIdx.x * blockDim.x + threadIdx.x;
    if (gid < n4) {
        float4 v = (gid < u4) ? ((const float4*)uf)[gid]
                              : ((const float4*)bf)[gid - u4];
        float4 o;
        o.x = 0.5f * v.x; o.y = 0.5f * v.y; o.z = 0.5f * v.z; o.w = 0.5f * v.w;
        ((float4*)acc)[gid] = o;
    }
    if (gid == 0) *loss = 0.0f;
}

// ---------------------------------------------------------------------------
// Kernel 2: scatter-add.  16 threads per edge; each thread: one float4 gather
// from embed_0[col], scale by 0.5*val, 4x global_atomic_add_f32 into acc[row].
// Entire 38.4MB accumulator is L2-resident on MI455X (192MB L2).
// ---------------------------------------------------------------------------
__global__ void k_scatter(const float* __restrict__ uf, const float* __restrict__ bf,
                          const float* __restrict__ vals,
                          const int* __restrict__ rows, const int* __restrict__ cols,
                          float* __restrict__ acc) {
    int gid = blockIdx.x * blockDim.x + threadIdx.x;   // NNZ*16 = 48,000,000 threads
    int e = gid >> 4;
    int p = gid & 15;                                  // which 4-float chunk of the row
    if (e >= NNZ) return;
    int c = cols[e];
    int r = rows[e];
    float w = 0.5f * vals[e];
    const float* src = (c < NUM_USERS) ? (uf + c * EMB)
                                       : (bf + (c - NUM_USERS) * EMB);
    float4 v = *(const float4*)(src + p * 4);
    float* dst = acc + r * EMB + p * 4;
    atomicAdd(dst + 0, w * v.x);
    atomicAdd(dst + 1, w * v.y);
    atomicAdd(dst + 2, w * v.z);
    atomicAdd(dst + 3, w * v.w);
}

// ---------------------------------------------------------------------------
// Kernel 3: loss = 1e-4 * sum(all_embeds^2).  16 elements/thread, block reduce.
// ---------------------------------------------------------------------------
__global__ void k_loss(const float* __restrict__ acc, float* __restrict__ loss) {
    __shared__ float red[256];
    const int n = N_NODES * EMB;                       // 9,600,000
    int base = (blockIdx.x * blockDim.x + threadIdx.x) * 16;
    float s = 0.0f;
    #pragma unroll
    for (int i = 0; i < 4; ++i) {
        int idx = base + i * 4;
        if (idx < n) {
            float4 v = *(const float4*)(acc + idx);
            s += v.x * v.x + v.y * v.y + v.z * v.z + v.w * v.w;
        }
    }
    red[threadIdx.x] = s;
    __syncthreads();
    for (int off = 128; off > 0; off >>= 1) {
        if (threadIdx.x < off) red[threadIdx.x] += red[threadIdx.x + off];
        __syncthreads();
    }
    if (threadIdx.x == 0) atomicAdd(loss, L2NORM * red[0]);
}

// ---------------------------------------------------------------------------
// Kernel 4: pred + user_score_bound via V_WMMA_F32_16X16X4_F32.
// One wave = (batch b, 16-candidate tile t).  2048*7 = 14336 waves exactly.
//   A (16x4 f32):  lanes 0-15  -> row m, K=0(v0)/K=1(v1)
//                  lanes 16-31 -> row m, K=2(v0)/K=3(v1)
//   B (4x16 f32):  user embedding broadcast to all 16 columns:
//                  v0: lanes0-15=u[k0], lanes16-31=u[k0+1]
//                  v1: lanes0-15=u[k0+2], lanes16-31=u[k0+3]
//   D: every column identical = dot(cand_row, u); lanes with N==0 (lane 0 and
//   lane 16) each hold 8 consecutive candidates' results in VGPRs 0..7.
// ---------------------------------------------------------------------------
__global__ void k_pred(const float* __restrict__ acc,
                       const int* __restrict__ users, const int* __restrict__ bundles,
                       const float* __restrict__ user_bound,
                       float* __restrict__ pred, float* __restrict__ usb) {
    const int lane = threadIdx.x & 31;
    const int wid  = blockIdx.x * (blockDim.x >> 5) + (threadIdx.x >> 5);
    const int b    = wid / 7;            // batch index [0, 2048)
    const int t    = wid % 7;            // candidate tile [0, 7), covers 112 >= 100
    const int half = lane >> 4;
    const int m    = lane & 15;

    const float* urow = acc + users[b] * EMB;

    int cidx = t * 16 + m;
    int cc   = cidx < NUM_CAND ? cidx : NUM_CAND - 1;   // clamp tail (store guarded)
    int bidx = bundles[b * NUM_CAND + cc];
    const float* arow = acc + (NUM_USERS + bidx) * EMB;

    v8f c = {};
    #pragma unroll
    for (int k0 = 0; k0 < EMB; k0 += 4) {
        v2f a = *(const v2f*)(arow + k0 + 2 * half);    // A[m][2h], A[m][2h+1]
        v2f bb;
        bb.x = urow[k0 + half];                          // B row (k0+half)
        bb.y = urow[k0 + 2 + half];                      // B row (k0+2+half)
        c = __builtin_amdgcn_wmma_f32_16x16x4_f32(
                /*neg_a=*/false, a, /*neg_b=*/false, bb,
                /*c_mod=*/(short)0, c, /*reuse_a=*/false, /*reuse_b=*/false);
    }

    // lanes with column N==0 write 8 consecutive candidates each
    if (m == 0) {
        int base = t * 16 + 8 * half;
        #pragma unroll
        for (int r = 0; r < 8; ++r) {
            int ci = base + r;
            if (ci < NUM_CAND) pred[b * NUM_CAND + ci] = c[r];
        }
    }

    // user_score_bound: one wave per batch (tile 0), plain wave reduction
    if (t == 0) {
        float p = urow[lane] * user_bound[lane] + urow[lane + 32] * user_bound[lane + 32];
        for (int off = 16; off > 0; off >>= 1) p += __shfl_xor(p, off, 32);
        if (lane == 0) usb[b] = p;
    }
}

// ---------------------------------------------------------------------------
extern "C" void kernel_launch(void* const* d_in, const int* in_sizes, int n_in,
                              void* d_out, int out_size, void* d_ws, size_t ws_size,
                              hipStream_t stream) {
    const float* uf      = (const float*)d_in[0];   // users_feature   100000x64
    const float* bf      = (const float*)d_in[1];   // bundles_feature  50000x64
    const float* ubound  = (const float*)d_in[2];   // user_bound       64x1
    const float* gvals   = (const float*)d_in[3];   // graph_vals       3M
    const int*   grows   = (const int*)  d_in[4];   // graph_rows       3M
    const int*   gcols   = (const int*)  d_in[5];   // graph_cols       3M
    const int*   users   = (const int*)  d_in[6];   // users            2048x1
    const int*   bundles = (const int*)  d_in[7];   // bundles          2048x100

    float* acc  = (float*)d_ws;                     // 150000x64 f32 = 38.4 MB
    float* pred = (float*)d_out;                    // [0, 204800)
    float* usb  = pred + BATCH * NUM_CAND;          // [204800, 206848)
    float* loss = usb + BATCH;                      // [206848]

    // 1) acc = 0.5 * embed_0 ; zero loss
    {
        int n4 = N_NODES * EMB / 4;
        k_init<<<(n4 + 255) / 256, 256, 0, stream>>>(uf, bf, acc, loss);
    }
    // 2) acc += 0.5 * graph_vals * embed_0[graph_cols] scattered to graph_rows
    {
        long long threads = (long long)NNZ * 16;    // 48,000,000
        k_scatter<<<(unsigned)((threads + 255) / 256), 256, 0, stream>>>(
            uf, bf, gvals, grows, gcols, acc);
    }
    // 3) loss reduction over acc
    {
        int nthreads = (N_NODES * EMB + 15) / 16;   // 600,000
        k_loss<<<(nthreads + 255) / 256, 256, 0, stream>>>(acc, loss);
    }
    // 4) pred + user_score_bound (WMMA f32 16x16x4)
    {
        int waves  = BATCH * 7;                     // 14336
        int blocks = waves / 8;                     // 256 thr = 8 waves -> 1792
        k_pred<<<blocks, 256, 0, stream>>>(acc, users, bundles, ubound, pred, usb);
    }
}